// OutBlock_14937896256222
// MI455X (gfx1250) — compile-verified
//
#include <hip/hip_runtime.h>
#include <hip/hip_bf16.h>

typedef __attribute__((ext_vector_type(2))) float v2f;
typedef __attribute__((ext_vector_type(8))) float v8f;

#define N_NODES 250000
#define C_IN    32
#define C_OUT   32
#define KNB     27
#define BN_EPS  1e-5f

#define TILES            (N_NODES / 16)       // 15625 exactly
#define NGROUPS          ((TILES + 1) / 2)    // 7813 tile-pairs
#define GEMM_BLOCKS      256
#define WAVES_PER_BLOCK  8
#define TOTAL_WAVES      (GEMM_BLOCKS * WAVES_PER_BLOCK)
#define STAT_BLOCKS      256

// ---------------------------------------------------------------------------
// Kernel 1: Y[N,32] = sum_k gather(x, neigh[:,k]) @ W[k]  via v_wmma_f32_16x16x4_f32
// Each wave: TWO 16-node tiles (4 x 16x16 f32 accumulators) so every LDS
// B-fragment pair feeds 4 WMMAs. W pre-packed in LDS as per-lane float2
// B-fragments (1 conflict-free ds_load_b64 per fragment).
// ---------------------------------------------------------------------------
__global__ void __launch_bounds__(256) octconv_wmma_kernel(
    const float* __restrict__ x, const int* __restrict__ neigh,
    const float* __restrict__ w, float* __restrict__ y)
{
    // [k][c(0..7)][h(0..1)] fragments, 32 lanes x float2 each: 13824*8B = 108 KB
    __shared__ v2f wfrag[KNB * 8 * 2 * 32];

    const int tid = threadIdx.x;

    // Pack weights into LDS in WMMA B-fragment layout:
    // lane L of fragment (k,c,h) = { W[k][4c+2*(L>>4)][16h+(L&15)], W[k][4c+2*(L>>4)+1][16h+(L&15)] }
    for (int e = tid; e < KNB * 8 * 2 * 32; e += 256) {
        int lane = e & 31;
        int h    = (e >> 5) & 1;
        int c    = (e >> 6) & 7;
        int k    = e >> 9;
        int kk   = 4 * c + 2 * (lane >> 4);
        int col  = 16 * h + (lane & 15);
        v2f wv;
        wv.x = w[(k * C_IN + kk) * C_OUT + col];
        wv.y = w[(k * C_IN + kk + 1) * C_OUT + col];
        wfrag[e] = wv;
    }
    __syncthreads();

    const int lane = tid & 31;
    const int wave = tid >> 5;
    const int koff = lane >> 4;   // 0/1: K-pair select (A) and M-half select (D)
    const int lrow = lane & 15;

    for (int g = blockIdx.x * WAVES_PER_BLOCK + wave; g < NGROUPS; g += TOTAL_WAVES) {
        const int t0 = 2 * g;
        const int t1 = min(2 * g + 1, TILES - 1);   // clamp: duplicate work, identical writes
        const int* nrow0 = neigh + (long)(t0 * 16 + lrow) * KNB;
        const int* nrow1 = neigh + (long)(t1 * 16 + lrow) * KNB;

        v8f a00 = {}, a01 = {};   // tile0: Cout 0..15 / 16..31
        v8f a10 = {}, a11 = {};   // tile1: Cout 0..15 / 16..31

        for (int k = 0; k < KNB; ++k) {
            const int idx0 = nrow0[k];                             // irregular gathers
            const int idx1 = nrow1[k];
            const float* xr0 = x + (long)idx0 * C_IN + 2 * koff;   // per-lane float2 (L2-resident)
            const float* xr1 = x + (long)idx1 * C_IN + 2 * koff;
            const v2f* bf = &wfrag[k * 16 * 32 + lane];
            #pragma unroll
            for (int c = 0; c < 8; ++c) {
                v2f b0 = bf[(2 * c + 0) * 32];                     // ds_load_b64, conflict-free
                v2f b1 = bf[(2 * c + 1) * 32];
                v2f aA = *(const v2f*)(xr0 + 4 * c);               // global_load_b64
                v2f aB = *(const v2f*)(xr1 + 4 * c);
                a00 = __builtin_amdgcn_wmma_f32_16x16x4_f32(false, aA, false, b0, (short)0, a00, false, false);
                a01 = __builtin_amdgcn_wmma_f32_16x16x4_f32(false, aA, false, b1, (short)0, a01, false, false);
                a10 = __builtin_amdgcn_wmma_f32_16x16x4_f32(false, aB, false, b0, (short)0, a10, false, false);
                a11 = __builtin_amdgcn_wmma_f32_16x16x4_f32(false, aB, false, b1, (short)0, a11, false, false);
            }
        }

        // D layout: VGPR v -> row base + v + 8*koff, col = 16*h + lrow
        float* yb0 = y + (long)t0 * 16 * C_OUT;
        float* yb1 = y + (long)t1 * 16 * C_OUT;
        #pragma unroll
        for (int v = 0; v < 8; ++v) {
            yb0[(v + 8 * koff) * C_OUT + lrow]      = a00[v];
            yb0[(v + 8 * koff) * C_OUT + 16 + lrow] = a01[v];
            yb1[(v + 8 * koff) * C_OUT + lrow]      = a10[v];
            yb1[(v + 8 * koff) * C_OUT + 16 + lrow] = a11[v];
        }
    }
}

// ---------------------------------------------------------------------------
// Kernel 2: deterministic per-block partial sum / sumsq per channel
// ---------------------------------------------------------------------------
__global__ void __launch_bounds__(256) stats_partial_kernel(
    const float* __restrict__ y, float* __restrict__ partials)
{
    __shared__ float ssum[256];
    __shared__ float ssq[256];
    const int tid = threadIdx.x;
    const int ch  = tid & 31;
    const int g   = tid >> 5;                                  // 0..7
    const int rowsPerBlock = (N_NODES + STAT_BLOCKS - 1) / STAT_BLOCKS;
    const int r0 = blockIdx.x * rowsPerBlock;
    const int r1 = min(r0 + rowsPerBlock, N_NODES);

    float s = 0.f, s2 = 0.f;
    for (int r = r0 + g; r < r1; r += 8) {                     // coalesced: 8 rows x 32 ch per pass
        float v = y[(long)r * C_OUT + ch];
        s += v; s2 += v * v;
    }
    ssum[tid] = s; ssq[tid] = s2;
    __syncthreads();
    for (int step = 128; step >= 32; step >>= 1) {
        if (tid < step) { ssum[tid] += ssum[tid + step]; ssq[tid] += ssq[tid + step]; }
        __syncthreads();
    }
    if (tid < 32) {
        partials[blockIdx.x * 64 + tid]      = ssum[tid];
        partials[blockIdx.x * 64 + 32 + tid] = ssq[tid];
    }
}

// ---------------------------------------------------------------------------
// Kernel 3: reduce partials in fixed order -> per-channel scale/bias
// ---------------------------------------------------------------------------
__global__ void __launch_bounds__(64) stats_finalize_kernel(
    const float* __restrict__ partials, const float* __restrict__ gamma,
    const float* __restrict__ beta, float* __restrict__ sb)
{
    const int t = threadIdx.x;                                 // 0..63
    float acc = 0.f;
    for (int b = 0; b < STAT_BLOCKS; ++b) acc += partials[b * 64 + t];
    __shared__ float tot[64];
    tot[t] = acc;
    __syncthreads();
    if (t < 32) {
        float mean  = tot[t] * (1.0f / N_NODES);
        float var   = tot[32 + t] * (1.0f / N_NODES) - mean * mean;
        float scale = gamma[t] * rsqrtf(var + BN_EPS);
        sb[t]      = scale;
        sb[32 + t] = beta[t] - mean * scale;
    }
}

// ---------------------------------------------------------------------------
// Kernel 4: in-place fused BN apply + ELU
// ---------------------------------------------------------------------------
__global__ void __launch_bounds__(256) bn_elu_kernel(
    float* __restrict__ y, const float* __restrict__ sb)
{
    __shared__ float sc[32], bi[32];
    if (threadIdx.x < 32)       sc[threadIdx.x]      = sb[threadIdx.x];
    else if (threadIdx.x < 64)  bi[threadIdx.x - 32] = sb[threadIdx.x];
    __syncthreads();

    const long total = (long)N_NODES * C_OUT;
    for (long i = (long)blockIdx.x * blockDim.x + threadIdx.x; i < total;
         i += (long)gridDim.x * blockDim.x) {
        int ch  = (int)(i & 31);
        float v = y[i] * sc[ch] + bi[ch];
        y[i] = v > 0.f ? v : expm1f(v);
    }
}

// ---------------------------------------------------------------------------
extern "C" void kernel_launch(void* const* d_in, const int* in_sizes, int n_in,
                              void* d_out, int out_size, void* d_ws, size_t ws_size,
                              hipStream_t stream) {
    const float* x      = (const float*)d_in[0];
    const int*   neigh  = (const int*)  d_in[1];
    const float* weight = (const float*)d_in[2];
    const float* gamma  = (const float*)d_in[3];
    const float* beta   = (const float*)d_in[4];
    float* y = (float*)d_out;

    float* partials = (float*)d_ws;                 // 256*64 floats
    float* sb       = partials + STAT_BLOCKS * 64;  // 64 floats (scale|bias)

    octconv_wmma_kernel<<<GEMM_BLOCKS, 256, 0, stream>>>(x, neigh, weight, y);
    stats_partial_kernel<<<STAT_BLOCKS, 256, 0, stream>>>(y, partials);
    stats_finalize_kernel<<<1, 64, 0, stream>>>(partials, gamma, beta, sb);
    bn_elu_kernel<<<2048, 256, 0, stream>>>(y, sb);
}